// EncoderDecoder_6657199309200
// MI455X (gfx1250) — compile-verified
//
#include <hip/hip_runtime.h>
#include <math.h>

// ---------------------------------------------------------------------------
// CDNA5 (gfx1250) fp32 WMMA path: V_WMMA_F32_16X16X4_F32, wave32.
// Weight panels staged LDS via the Tensor Data Mover (tensor_load_to_lds,
// TENSORcnt). Operand layouts (ISA 7.12.2):
//   A (16x4):  lane l: m=l&15, a[i] = A[m][2*(l>>4)+i]
//   B (4x16):  lane l: n=l&15, b[i] = B[2*(l>>4)+i][n]
//   C/D(16x16):lane l: n=l&15, d[j] = C[j+8*(l>>4)][n]
// ---------------------------------------------------------------------------

typedef float v2f __attribute__((ext_vector_type(2)));
typedef float v8f __attribute__((ext_vector_type(8)));
typedef unsigned int v4u __attribute__((ext_vector_type(4)));
typedef int v4i __attribute__((ext_vector_type(4)));
typedef int v8i __attribute__((ext_vector_type(8)));

#define WMMA_F32(a, b, c) \
  __builtin_amdgcn_wmma_f32_16x16x4_f32(false, (a), false, (b), (short)0, (c), false, false)

constexpr int Hd  = 256;
constexpr int NHd = 8;
constexpr int DHd = 32;
constexpr int FFd = 1024;
constexpr int Bd  = 2;
constexpr int Ld  = 4096;        // T*N
constexpr int BLd = Bd * Ld;

// ---- TDM: DMA a (rows x 16) f32 tile (row-major, row stride = row_len) into LDS
// D# built per ISA 8.3/8.4: group0 = {count|flags, lds_addr, global_addr, type=2},
// group1 = {data_size, tensor_dim0/1, tile_dim0/1, tensor_dim0_stride}.
// This toolchain exposes the 6-arg builtin: (g0, g1, g2, g3, extra, cpol).
__device__ __forceinline__ void tdm_load_panel_f32(unsigned int lds_off, const float* gptr,
                                                   unsigned int rows, unsigned int row_len) {
  unsigned long long ga = (unsigned long long)(uintptr_t)gptr;
  v4u g0;
  g0[0] = 1u;                                       // count=1, user descriptor
  g0[1] = lds_off;                                  // lds_addr (bytes)
  g0[2] = (unsigned int)(ga & 0xffffffffull);       // global_addr[31:0]
  g0[3] = (unsigned int)(ga >> 32) | (2u << 30);    // global_addr[56:32] | type=2
  v8i g1;
  unsigned int td0 = row_len;                       // tensor_dim0 (elems per row)
  unsigned int td1 = rows;                          // tensor_dim1
  g1[0] = (int)(2u << 16);                          // data_size=2 (4 bytes)
  g1[1] = (int)((td0 & 0xffffu) << 16);             // tensor_dim0[15:0] @ bits 63:48
  g1[2] = (int)((td0 >> 16) | ((td1 & 0xffffu) << 16));
  g1[3] = (int)((td1 >> 16) | (16u << 16));         // tile_dim0=16 @ bits 127:112
  g1[4] = (int)(rows & 0xffffu);                    // tile_dim1=rows, tile_dim2=0
  g1[5] = (int)row_len;                             // tensor_dim0_stride[31:0]
  g1[6] = 0;                                        // stride0[47:32] | stride1[15:0]
  g1[7] = 0;                                        // stride1[47:16]
  v4i gz4 = {0, 0, 0, 0};
  v8i gz8 = {0, 0, 0, 0, 0, 0, 0, 0};
  __builtin_amdgcn_tensor_load_to_lds(g0, g1, gz4, gz4, gz8, 0);
}

// --------------------------- pack (stable, serial per batch) ----------------
__global__ void pack_kernel(const float* __restrict__ x, float* __restrict__ conds,
                            int* __restrict__ counts) {
  int b = blockIdx.x;
  if (threadIdx.x != 0) return;
  const float* ch0 = x + (size_t)b * 3 * Ld;
  const float* tg  = ch0 + Ld;
  const float* xg  = ch0 + 2 * Ld;
  int cnt = 0;
  for (int i = 0; i < Ld; ++i) {
    if (ch0[i] != -1.0f) {
      size_t o = ((size_t)b * Ld + cnt) * 3;
      conds[o + 0] = ch0[i];
      conds[o + 1] = tg[i];
      conds[o + 2] = xg[i];
      ++cnt;
    }
  }
  for (int i = cnt; i < Ld; ++i) {
    size_t o = ((size_t)b * Ld + i) * 3;
    conds[o + 0] = 0.0f; conds[o + 1] = 0.0f; conds[o + 2] = 0.0f;
  }
  counts[b] = cnt;
}

// --------------------------- embeddings ------------------------------------
__global__ void enc_embed_kernel(const float* __restrict__ conds, const float* __restrict__ W,
                                 const float* __restrict__ bias, float* __restrict__ out) {
  int r = blockIdx.x, t = threadIdx.x;
  float acc = bias[t];
#pragma unroll
  for (int c = 0; c < 3; ++c) acc += conds[(size_t)r * 3 + c] * W[c * Hd + t];
  out[(size_t)r * Hd + t] = acc;
}

__global__ void dec_embed_kernel(const float* __restrict__ x, const float* __restrict__ W,
                                 const float* __restrict__ bias, float* __restrict__ out) {
  int r = blockIdx.x, t = threadIdx.x;
  int b = r / Ld, i = r - b * Ld;
  float tg = x[((size_t)b * 3 + 1) * Ld + i];
  float xg = x[((size_t)b * 3 + 2) * Ld + i];
  out[(size_t)r * Hd + t] = bias[t] + tg * W[t] + xg * W[Hd + t];
}

// --------------------------- WMMA GEMM with TDM-staged weights --------------
// C[M,N] = A[M,K] @ W[K,N] + bias ; optional ReLU.
// One block = 4 waves; block owns one 16-col weight panel (TDM -> LDS once),
// each wave computes one of 4 consecutive 16-row M tiles against it.
// grid.x = (M/64) * (N/16); dynamic LDS = K*16*4 bytes. M % 64 == 0 here.
__global__ void gemm_kernel(const float* __restrict__ A, const float* __restrict__ W,
                            const float* __restrict__ bias, float* __restrict__ C,
                            int M, int N, int K, int relu) {
  extern __shared__ float lwb[];          // K x 16 weight panel
  int ntiles = N >> 4;
  int mg = blockIdx.x / ntiles;
  int nt = blockIdx.x - mg * ntiles;
  int wv = threadIdx.x >> 5;
  int mt = mg * 4 + wv;
  int lane = threadIdx.x & 31;
  int nl = lane & 15, half = lane >> 4;

  if (wv == 0) {                          // wave-uniform: one TDM DMA per block
    tdm_load_panel_f32((unsigned int)(uintptr_t)lwb, W + nt * 16,
                       (unsigned int)K, (unsigned int)N);
    __builtin_amdgcn_s_wait_tensorcnt(0);
  }
  __syncthreads();                        // publish LDS panel to all 4 waves

  const float* Arow = A + (size_t)(mt * 16 + nl) * K;
  v8f acc = {0.f, 0.f, 0.f, 0.f, 0.f, 0.f, 0.f, 0.f};
  for (int c = 0; c < K; c += 4) {
    int cc = c + 2 * half;
    v2f a;  a.x = Arow[cc];             a.y = Arow[cc + 1];
    v2f bv; bv.x = lwb[cc * 16 + nl];   bv.y = lwb[(cc + 1) * 16 + nl];
    acc = WMMA_F32(a, bv, acc);
  }
  float bb = bias[nt * 16 + nl];
#pragma unroll
  for (int j = 0; j < 8; ++j) {
    float val = acc[j] + bb;
    if (relu) val = fmaxf(val, 0.0f);
    C[(size_t)(mt * 16 + j + 8 * half) * N + nt * 16 + nl] = val;
  }
}

// --------------------------- flash attention (one wave / 16 queries) --------
// Q,K,V,O buffers: (B*L, H) row-major; head h occupies cols [h*DH, h*DH+32).
// S_T = K_tile(16k x 32d) @ Q^T(32d x 16q);  O^T = V^T(32d x 16k) @ P_T(16k x 16q)
__global__ void attn_kernel(const float* __restrict__ qb, const float* __restrict__ kb,
                            const float* __restrict__ vb, float* __restrict__ ob,
                            const int* __restrict__ counts) {
  const int QT = Ld / 16;
  int wid = blockIdx.x;
  int qt = wid % QT;
  int h  = (wid / QT) % NHd;
  int b  = wid / (QT * NHd);
  int lane = threadIdx.x & 31;
  int nl = lane & 15, half = lane >> 4;
  int q0 = qt * 16;
  int cnt = counts[b];
  const float scale = 0.17677669529663687f;   // 1/sqrt(32)

  // Preload Q tile in B-operand layout: qr[t] = Q[q0+nl][4t+2*half .. +1]
  v2f qr[8];
  {
    const float* Qrow = qb + (size_t)(b * Ld + q0 + nl) * Hd + h * DHd;
#pragma unroll
    for (int t = 0; t < 8; ++t) {
      int c = 4 * t + 2 * half;
      qr[t].x = Qrow[c]; qr[t].y = Qrow[c + 1];
    }
  }

  v8f olo = {0.f, 0.f, 0.f, 0.f, 0.f, 0.f, 0.f, 0.f};
  v8f ohi = {0.f, 0.f, 0.f, 0.f, 0.f, 0.f, 0.f, 0.f};
  float run_max = -__builtin_inff();
  float run_sum = 0.0f;

  for (int kt = 0; kt < Ld / 16; ++kt) {
    int k0 = kt * 16;
    // ---- S_T = K @ Q^T : A = K rows (m = key), contraction over DH=32 ----
    const float* Krow = kb + (size_t)(b * Ld + k0 + nl) * Hd + h * DHd;
    v8f s = {0.f, 0.f, 0.f, 0.f, 0.f, 0.f, 0.f, 0.f};
#pragma unroll
    for (int t = 0; t < 8; ++t) {
      int c = 4 * t + 2 * half;
      v2f ka; ka.x = Krow[c]; ka.y = Krow[c + 1];
      s = WMMA_F32(ka, qr[t], s);
    }
    // s[j] = score(query q0+nl, key k0 + j + 8*half)
    float sm[8], p[8];
    float mx = -__builtin_inff();
#pragma unroll
    for (int j = 0; j < 8; ++j) {
      int key = k0 + j + 8 * half;
      sm[j] = (key < cnt) ? s[j] * scale : -1.0e9f;
      mx = fmaxf(mx, sm[j]);
    }
    mx = fmaxf(mx, __shfl_xor(mx, 16, 32));      // partner lane holds other 8 keys
    float nm   = fmaxf(run_max, mx);
    float corr = __expf(run_max - nm);
    float psum = 0.0f;
#pragma unroll
    for (int j = 0; j < 8; ++j) { p[j] = __expf(sm[j] - nm); psum += p[j]; }
    psum   += __shfl_xor(psum, 16, 32);
    run_sum = run_sum * corr + psum;
    run_max = nm;
    olo *= corr;
    ohi *= corr;

    // swap halves so every lane can build the B operand of P_T
    float ps[8];
#pragma unroll
    for (int j = 0; j < 8; ++j) ps[j] = __shfl_xor(p[j], 16, 32);

    // ---- O^T += V^T @ P_T : contraction over 16 keys (4 wmma steps) ----
    const float* Vb = vb + (size_t)(b * Ld + k0) * Hd + h * DHd;
#pragma unroll
    for (int t = 0; t < 4; ++t) {
      const int base = 4 * (t & 1);
      const int s1   = t >> 1;     // compile-time after unroll
      float a0 = s1 ? ps[base + 0] : p[base + 0];
      float a1 = s1 ? ps[base + 1] : p[base + 1];
      float b0 = s1 ? p[base + 2]  : ps[base + 2];
      float b1 = s1 ? p[base + 3]  : ps[base + 3];
      v2f bb; bb.x = half ? b0 : a0; bb.y = half ? b1 : a1;

      int c = 4 * t + 2 * half;    // key index of a[i=0]
      v2f alo, ahi;
      alo.x = Vb[(size_t)(c + 0) * Hd + nl];
      alo.y = Vb[(size_t)(c + 1) * Hd + nl];
      ahi.x = Vb[(size_t)(c + 0) * Hd + 16 + nl];
      ahi.y = Vb[(size_t)(c + 1) * Hd + 16 + nl];
      olo = WMMA_F32(alo, bb, olo);
      ohi = WMMA_F32(ahi, bb, ohi);
    }
  }

  float inv = 1.0f / run_sum;
  // O^T layout: lane nl = query col, row j+8*half = channel d (lo: 0..15, hi: 16..31)
  float* Orow = ob + (size_t)(b * Ld + q0 + nl) * Hd + h * DHd;
#pragma unroll
  for (int j = 0; j < 8; ++j) {
    Orow[j + 8 * half]      = olo[j] * inv;
    Orow[16 + j + 8 * half] = ohi[j] * inv;
  }
}

// --------------------------- residual + LayerNorm ---------------------------
__global__ void ln_kernel(const float* __restrict__ x, const float* __restrict__ y,
                          const float* __restrict__ g, const float* __restrict__ be,
                          float* __restrict__ z) {
  int r = blockIdx.x, t = threadIdx.x;
  size_t base = (size_t)r * Hd;
  float v = x[base + t] + y[base + t];
  __shared__ float red[Hd];
  red[t] = v; __syncthreads();
  for (int s = Hd / 2; s > 0; s >>= 1) { if (t < s) red[t] += red[t + s]; __syncthreads(); }
  float mu = red[0] * (1.0f / Hd);
  __syncthreads();
  float dv = v - mu;
  red[t] = dv * dv; __syncthreads();
  for (int s = Hd / 2; s > 0; s >>= 1) { if (t < s) red[t] += red[t + s]; __syncthreads(); }
  float var = red[0] * (1.0f / Hd);
  float rs  = rsqrtf(var + 1e-5f);
  z[base + t] = dv * rs * g[t] + be[t];
}

// --------------------------- head (H -> 1) ----------------------------------
__global__ void head_kernel(const float* __restrict__ q, const float* __restrict__ w,
                            const float* __restrict__ bsc, float* __restrict__ out) {
  int r = blockIdx.x, t = threadIdx.x;
  __shared__ float red[Hd];
  red[t] = q[(size_t)r * Hd + t] * w[t];
  __syncthreads();
  for (int s = Hd / 2; s > 0; s >>= 1) { if (t < s) red[t] += red[t + s]; __syncthreads(); }
  if (t == 0) out[r] = red[0] + bsc[0];
}

// ---------------------------------------------------------------------------
extern "C" void kernel_launch(void* const* d_in, const int* in_sizes, int n_in,
                              void* d_out, int out_size, void* d_ws, size_t ws_size,
                              hipStream_t stream) {
  (void)in_sizes; (void)n_in; (void)out_size; (void)ws_size;
  auto P = [&](int i) { return (const float*)d_in[i]; };
  // d_in order: x, enc_embed{w,b}, enc_layers[2]x16, dec_embed{w,b}, dec_layers[2]x16, head{w,b}
  const float* x   = P(0);
  const float* eew = P(1);  const float* eeb = P(2);
  const float* dew = P(35); const float* deb = P(36);
  const float* hw  = P(69); const float* hb  = P(70);

  float* ws    = (float*)d_ws;
  float* conds = ws;                          // B*L*3
  int*   counts = (int*)(ws + (size_t)Bd * Ld * 3);
  const size_t o0 = 32768;
  const size_t SL = (size_t)BLd * Hd;
  float* Henc = ws + o0;                      // encoder stream / output
  float* Xs   = ws + o0 + 1 * SL;             // decoder stream
  float* Qb   = ws + o0 + 2 * SL;
  float* Kb   = ws + o0 + 3 * SL;
  float* Vb   = ws + o0 + 4 * SL;
  float* Yb   = ws + o0 + 5 * SL;             // attention heads out
  float* FFb  = ws + o0 + 6 * SL;             // B*L*FF (= 4*SL)

  dim3 b32(32), b128(128), b256(Hd);

  auto gemm = [&](const float* A, const float* Wm, const float* bias, float* C,
                  int M, int Nn, int K, int relu) {
    int blocks = (M / 64) * (Nn / 16);         // 4 waves/block, 4 M-tiles/block
    size_t lds = (size_t)K * 16 * sizeof(float);
    gemm_kernel<<<dim3(blocks), b128, lds, stream>>>(A, Wm, bias, C, M, Nn, K, relu);
  };

  auto block_layer = [&](float* S, const float* KV, int pb) {
    // param order inside a block: wq,bq,wk,bk,wv,bv,wo,bo,ln1_g,ln1_b,w1,b1,w2,b2,ln2_g,ln2_b
    gemm(S,  P(pb + 0), P(pb + 1), Qb, BLd, Hd, Hd, 0);
    gemm(KV, P(pb + 2), P(pb + 3), Kb, BLd, Hd, Hd, 0);
    gemm(KV, P(pb + 4), P(pb + 5), Vb, BLd, Hd, Hd, 0);
    attn_kernel<<<dim3(Bd * NHd * (Ld / 16)), b32, 0, stream>>>(Qb, Kb, Vb, Yb, counts);
    gemm(Yb, P(pb + 6), P(pb + 7), Qb, BLd, Hd, Hd, 0);
    ln_kernel<<<dim3(BLd), b256, 0, stream>>>(S, Qb, P(pb + 8), P(pb + 9), S);
    gemm(S,   P(pb + 10), P(pb + 11), FFb, BLd, FFd, Hd, 1);
    gemm(FFb, P(pb + 12), P(pb + 13), Qb,  BLd, Hd, FFd, 0);
    ln_kernel<<<dim3(BLd), b256, 0, stream>>>(S, Qb, P(pb + 14), P(pb + 15), S);
  };

  pack_kernel<<<dim3(Bd), dim3(1), 0, stream>>>(x, conds, counts);
  enc_embed_kernel<<<dim3(BLd), b256, 0, stream>>>(conds, eew, eeb, Henc);
  block_layer(Henc, Henc, 3);    // encoder layer 0 (self-attn, masked keys)
  block_layer(Henc, Henc, 19);   // encoder layer 1
  dec_embed_kernel<<<dim3(BLd), b256, 0, stream>>>(x, dew, deb, Xs);
  block_layer(Xs, Henc, 37);     // decoder layer 0 (cross-attn on encoder out)
  block_layer(Xs, Henc, 53);     // decoder layer 1
  head_kernel<<<dim3(BLd), b256, 0, stream>>>(Xs, hw, hb, (float*)d_out);
}